// CenterWoParamMultiCosineSoftmaxLoss_15917148799628
// MI455X (gfx1250) — compile-verified
//
#include <hip/hip_runtime.h>

// ---------------- problem constants (match reference) ----------------
#define B_N   16384
#define D_DIM 512
#define C_N   90
#define K_N   32
#define MAX_TILES (C_N + B_N / 16)   // 1114 worst case: every class has a partial tile

typedef __attribute__((ext_vector_type(2))) float v2f;
typedef __attribute__((ext_vector_type(8))) float v8f;

// ---------------- workspace layout ----------------
struct Ws {
    int   counts[128];
    int   class_start[128];      // prefix sums; class_start[c+1] = end of class c
    int   cursor[128];
    int   tile_class[MAX_TILES];
    int   tile_base[MAX_TILES];  // global index into order[]
    int   num_tiles;
    int   pad_;
    int   order[B_N];
    float per_sample[B_N];
    float inv_cnorm[C_N * K_N];
};

// ---------------- tiny setup kernels ----------------
__global__ void k_zero(Ws* ws) {
    int t = threadIdx.x;
    if (t < 128) { ws->counts[t] = 0; ws->cursor[t] = 0; }
}

// inverse L2 norms of the 2880 center rows (one wave per row)
__global__ void k_cnorm(const float* __restrict__ centers, Ws* __restrict__ ws) {
    int row  = blockIdx.x * 8 + (threadIdx.x >> 5);   // 360 blocks * 8 waves = 2880 rows
    int lane = threadIdx.x & 31;
    const float* p = centers + (size_t)row * D_DIM;
    float s = 0.f;
    for (int j = lane; j < D_DIM; j += 32) { float v = p[j]; s += v * v; }
    for (int off = 16; off; off >>= 1) s += __shfl_xor(s, off, 32);
    if (lane == 0) ws->inv_cnorm[row] = rsqrtf(s + 1e-12f);
}

__global__ void k_count(const int* __restrict__ labels, Ws* __restrict__ ws) {
    int t = blockIdx.x * blockDim.x + threadIdx.x;
    if (t < B_N) atomicAdd(&ws->counts[labels[t]], 1);
}

// serial scan + tile-map build (tiny: 90 classes, <=1114 tiles)
__global__ void k_scan(Ws* ws) {
    if (threadIdx.x != 0 || blockIdx.x != 0) return;
    int total = 0;
    for (int c = 0; c < C_N; ++c) { ws->class_start[c] = total; total += ws->counts[c]; }
    ws->class_start[C_N] = total;
    int nt = 0;
    for (int c = 0; c < C_N; ++c) {
        int cnt = ws->counts[c];
        for (int b = 0; b < cnt; b += 16) {
            ws->tile_class[nt] = c;
            ws->tile_base[nt]  = ws->class_start[c] + b;
            ++nt;
        }
    }
    ws->num_tiles = nt;
}

__global__ void k_scatter(const int* __restrict__ labels, Ws* __restrict__ ws) {
    int t = blockIdx.x * blockDim.x + threadIdx.x;
    if (t < B_N) {
        int l   = labels[t];
        int pos = atomicAdd(&ws->cursor[l], 1);
        ws->order[ws->class_start[l] + pos] = t;
    }
}

// ---------------- main WMMA kernel ----------------
// One block (2 waves) per tile of 16 same-class samples.
// Wave w computes the 16x16 tile of raw dots against sub-centers [16w, 16w+16)
// with v_wmma_f32_16x16x4_f32, contracting over D=512 in 128 steps.
#define XS_STRIDE 516   // 516*4 B rows => column reads of A-fragments hit distinct LDS banks

__global__ __launch_bounds__(64) void k_main(const float* __restrict__ x,
                                             const float* __restrict__ centers,
                                             Ws* __restrict__ ws) {
    __shared__ float Xs[16][XS_STRIDE];
    __shared__ float dstS[16][K_N];
    __shared__ float invX[16];
    __shared__ float partial[16][4];
    __shared__ int   sidx[16];
    __shared__ int   sCls, sBase, sEnd;

    const int t = blockIdx.x;
    if (t >= ws->num_tiles) return;            // uniform branch: EXEC stays all-ones
    const int tid = threadIdx.x;

    if (tid == 0) {
        int c  = ws->tile_class[t];
        sCls   = c;
        sBase  = ws->tile_base[t];
        sEnd   = ws->class_start[c + 1];
    }
    __syncthreads();
    const int cls = sCls, base = sBase, cend = sEnd;

    if (tid < 16) {
        int i = base + tid;
        sidx[tid] = ws->order[i < cend ? i : (cend - 1)];   // pad rows duplicate last valid
    }
    __syncthreads();

    // ---- stage 16 x-rows into LDS (coalesced float4 per row) ----
    for (int m = 0; m < 16; ++m) {
        const float* src = x + (size_t)sidx[m] * D_DIM;
        for (int c = tid * 4; c < D_DIM; c += 64 * 4) {
            float4 v = *(const float4*)(src + c);
            *(float4*)(&Xs[m][c]) = v;
        }
    }
    __syncthreads();

    // ---- per-sample inverse norms from the staged tile ----
    {
        int m = tid & 15, q = tid >> 4;
        float s = 0.f;
        const int c0 = q * 128;
        for (int c = c0; c < c0 + 128; ++c) { float v = Xs[m][c]; s += v * v; }
        partial[m][q] = s;
    }
    __syncthreads();
    if (tid < 16) {
        float s = partial[tid][0] + partial[tid][1] + partial[tid][2] + partial[tid][3];
        invX[tid] = rsqrtf(s + 1e-12f);
    }
    __syncthreads();

    // ---- GEMM: 16 samples x 16 sub-centers per wave, K = 512 ----
    const int w    = tid >> 5;
    const int lane = tid & 31;
    const int mrow = lane & 15;          // A: M row | B: N column
    const int kk   = (lane >> 4) * 2;    // f32 16x16x4 fragment: v0/v1 hold K pair (0,1) or (2,3)

    const float* cenW = centers + ((size_t)cls * K_N + w * 16 + mrow) * D_DIM + kk;
    const float* xRow = &Xs[mrow][kk];

    v8f acc = {};
    #pragma unroll 8
    for (int d = 0; d < D_DIM; d += 4) {
        v2f a = *(const v2f*)(xRow + d);   // ds_load_b64
        v2f b = *(const v2f*)(cenW + d);   // global_load_b64 (L2-resident centers)
        acc = __builtin_amdgcn_wmma_f32_16x16x4_f32(
                  false, a, false, b, (short)0, acc, false, false);
    }

    // ---- fold in normalization, write dst = 1 - cos into LDS ----
    const int ng   = w * 16 + mrow;                    // global sub-center index
    const float invc = ws->inv_cnorm[cls * K_N + ng];
    const int mAdd = (lane >> 4) * 8;                  // C/D layout: vgpr r -> M=r (+8 for hi lanes)
    #pragma unroll
    for (int r = 0; r < 8; ++r) {
        int m = r + mAdd;
        float cosv = acc[r] * invX[m] * invc;
        dstS[m][ng] = 1.0f - cosv;
    }
    __syncthreads();

    // ---- per-sample softmax over 32 sub-centers ----
    if (tid < 16) {
        int i = base + tid;
        if (i < cend) {
            float mx = -1e30f;
            #pragma unroll
            for (int k = 0; k < K_N; ++k) {
                float v = 2.0f - dstS[tid][k];
                mx = fmaxf(mx, v);
            }
            float se = 0.f, sw = 0.f;
            #pragma unroll
            for (int k = 0; k < K_N; ++k) {
                float dv = dstS[tid][k];
                float e  = expf((2.0f - dv) - mx);
                se += e;
                sw += e * dv;
            }
            ws->per_sample[ws->order[i]] = sw / se;
        }
    }
}

// ---------------- deterministic final mean ----------------
__global__ void k_reduce(const Ws* __restrict__ ws, float* __restrict__ out) {
    __shared__ float sh[256];
    float s = 0.f;
    for (int i = threadIdx.x; i < B_N; i += 256) s += ws->per_sample[i];
    sh[threadIdx.x] = s;
    __syncthreads();
    for (int off = 128; off; off >>= 1) {
        if (threadIdx.x < off) sh[threadIdx.x] += sh[threadIdx.x + off];
        __syncthreads();
    }
    if (threadIdx.x == 0) out[0] = sh[0] * (1.0f / (float)B_N);
}

// ---------------- host-side launch ----------------
extern "C" void kernel_launch(void* const* d_in, const int* in_sizes, int n_in,
                              void* d_out, int out_size, void* d_ws, size_t ws_size,
                              hipStream_t stream) {
    const float* x       = (const float*)d_in[0];
    const int*   labels  = (const int*)d_in[1];
    const float* centers = (const float*)d_in[2];
    float*       out     = (float*)d_out;
    Ws*          ws      = (Ws*)d_ws;

    k_zero   <<<1, 128, 0, stream>>>(ws);
    k_cnorm  <<<(C_N * K_N) / 8, 256, 0, stream>>>(centers, ws);
    k_count  <<<B_N / 256, 256, 0, stream>>>(labels, ws);
    k_scan   <<<1, 1, 0, stream>>>(ws);
    k_scatter<<<B_N / 256, 256, 0, stream>>>(labels, ws);
    k_main   <<<MAX_TILES, 64, 0, stream>>>(x, centers, ws);
    k_reduce <<<1, 256, 0, stream>>>(ws, out);
}